// VTBPR_84275848282700
// MI455X (gfx1250) — compile-verified
//
#include <hip/hip_runtime.h>
#include <hip/hip_bf16.h>

// ---------------------------------------------------------------------------
// VT-BPR forward on MI455X (gfx1250).
//
// Roofline: out = 256 MB metapath broadcast + 32 KB scores; reads ~100 MB.
// ~25 MFLOP total -> pure HBM-bandwidth problem (~15 us at 23.3 TB/s).
//   * metapath: 1 block / batch row; load gathered ug/ig rows ONCE into
//     registers (float4 per thread), replicate 16x with b128 NT stores.
//   * score: 3 batched row-dots done as diag(A x B^T) via
//     V_WMMA_F32_16X16X4_F32, accumulated over 128 K=4 chunks. A(16x4) and
//     B(4x16) fragments share the same lane layout -> both are float2 loads.
// ---------------------------------------------------------------------------

typedef __attribute__((ext_vector_type(2))) float v2f;
typedef __attribute__((ext_vector_type(4))) float v4f;
typedef __attribute__((ext_vector_type(8))) float v8f;

#define HIDDEN 512
#define PL 16  // P*L = 4*4

// ------------------------- score kernel (WMMA) -----------------------------
// One wave handles a tile of 16 batch rows. For each K=4 chunk:
//   D[i][j] += sum_k A[i][k] * B[k][j],  B[k][j] = partner_row_j[4c+k]
// so diag(D) = per-row dot products. Three products share one accumulator.
__global__ __launch_bounds__(128) void vtbpr_score_kernel(
    const float* __restrict__ user_gama,
    const float* __restrict__ item_gama,
    const float* __restrict__ user_beta,
    const float* __restrict__ item_beta,
    const float* __restrict__ theta_v,
    const float* __restrict__ theta_t,
    const float* __restrict__ vis,
    const float* __restrict__ txt,
    const int*   __restrict__ user_idx,
    const int*   __restrict__ item_idx,
    float*       __restrict__ score)
{
    const int lane = threadIdx.x & 31;
    const int wave = threadIdx.x >> 5;
    const int tile = blockIdx.x * (blockDim.x >> 5) + wave;  // 16 rows / tile
    const int m    = lane & 15;                              // row within tile
    const int row  = tile * 16 + m;                          // batch row
    const int koff = (lane < 16) ? 0 : 2;                    // K sub-offset

    const int ui = user_idx[row];
    const int ii = item_idx[row];

    const float* ug = user_gama + (size_t)ui  * HIDDEN;
    const float* ig = item_gama + (size_t)ii  * HIDDEN;
    const float* tv = theta_v   + (size_t)ui  * HIDDEN;
    const float* tt = theta_t   + (size_t)ui  * HIDDEN;
    const float* vf = vis       + (size_t)row * HIDDEN;
    const float* tf = txt       + (size_t)row * HIDDEN;

    v8f acc = {0.f, 0.f, 0.f, 0.f, 0.f, 0.f, 0.f, 0.f};

    #pragma unroll 4
    for (int c = 0; c < HIDDEN / 4; ++c) {
        const int o = c * 4 + koff;
        // A(16x4) and B(4x16) fragments: lane p holds {x[4c+koff], x[4c+koff+1]}
        v2f a0 = *(const v2f*)(ug + o);
        v2f b0 = *(const v2f*)(ig + o);
        v2f a1 = *(const v2f*)(tv + o);
        v2f b1 = *(const v2f*)(vf + o);
        v2f a2 = *(const v2f*)(tt + o);
        v2f b2 = *(const v2f*)(tf + o);
        acc = __builtin_amdgcn_wmma_f32_16x16x4_f32(false, a0, false, b0,
                                                    (short)0, acc, false, false);
        acc = __builtin_amdgcn_wmma_f32_16x16x4_f32(false, a1, false, b1,
                                                    (short)0, acc, false, false);
        acc = __builtin_amdgcn_wmma_f32_16x16x4_f32(false, a2, false, b2,
                                                    (short)0, acc, false, false);
    }

    // Diagonal extraction (C/D layout: lanes 0-15 -> M=0..7 in V0..7,
    // lanes 16-31 -> M=8..15 in V0..7, N = lane&15).
    // D[i][i]: i<8 -> lane i, vgpr i;  i>=8 -> lane i+16, vgpr i-8.
    const bool has  = (lane < 8) || (lane >= 24);
    const int  vidx = (lane < 8) ? lane : (lane - 24);  // 0..7
    const int  orow = (lane < 8) ? lane : (lane - 16);  // row within tile

    float d = (vidx & 4)
        ? ((vidx & 2) ? ((vidx & 1) ? acc[7] : acc[6])
                      : ((vidx & 1) ? acc[5] : acc[4]))
        : ((vidx & 2) ? ((vidx & 1) ? acc[3] : acc[2])
                      : ((vidx & 1) ? acc[1] : acc[0]));

    if (has) {
        const int gr = tile * 16 + orow;
        const float ub = user_beta[user_idx[gr]];
        const float ib = item_beta[item_idx[gr]];
        score[gr] = ub + ib + d;
    }
}

// ----------------------- metapath broadcast kernel -------------------------
// One block per batch row b. 128 threads x float4 = the full 512-float row.
// Load gathered ug/ig rows once into registers, then write 16 selected
// copies with non-temporal b128 stores (write-once 256 MB stream).
__global__ __launch_bounds__(128) void vtbpr_metapath_kernel(
    const float* __restrict__ user_gama,
    const float* __restrict__ item_gama,
    const int*   __restrict__ user_idx,
    const int*   __restrict__ item_idx,
    const int*   __restrict__ path_type,
    float*       __restrict__ out)   // points at metapath region
{
    const int b = blockIdx.x;
    const int t = threadIdx.x;       // 0..127, handles floats [4t, 4t+3]

    const int ui = user_idx[b];
    const int ii = item_idx[b];

    const v4f ug = *(const v4f*)(user_gama + (size_t)ui * HIDDEN + t * 4);
    const v4f ig = *(const v4f*)(item_gama + (size_t)ii * HIDDEN + t * 4);
    const v4f zz = {0.f, 0.f, 0.f, 0.f};

    float* base = out + ((size_t)b * PL) * HIDDEN + t * 4;

    #pragma unroll
    for (int pl = 0; pl < PL; ++pl) {
        const int tp = path_type[b * PL + pl];      // uniform -> scalar load
        const v4f v  = (tp == 0) ? ug : ((tp == 1) ? ig : zz);
        __builtin_nontemporal_store(v, (v4f*)(base + pl * HIDDEN));
    }
}

// ------------------------------- launcher ----------------------------------
extern "C" void kernel_launch(void* const* d_in, const int* in_sizes, int n_in,
                              void* d_out, int out_size, void* d_ws, size_t ws_size,
                              hipStream_t stream)
{
    const float* user_gama = (const float*)d_in[0];
    const float* item_gama = (const float*)d_in[1];
    const float* user_beta = (const float*)d_in[2];
    const float* item_beta = (const float*)d_in[3];
    const float* theta_v   = (const float*)d_in[4];
    const float* theta_t   = (const float*)d_in[5];
    const float* vis       = (const float*)d_in[6];
    const float* txt       = (const float*)d_in[7];
    const int*   user_idx  = (const int*)d_in[8];
    const int*   item_idx  = (const int*)d_in[9];
    const int*   path_type = (const int*)d_in[10];

    float* out = (float*)d_out;
    const int B = in_sizes[8];                 // 8192

    // Score: 16 rows per wave, 4 waves per block of 128.
    const int tiles = B / 16;                  // 512
    const int score_blocks = tiles / 4;        // 128
    vtbpr_score_kernel<<<score_blocks, 128, 0, stream>>>(
        user_gama, item_gama, user_beta, item_beta,
        theta_v, theta_t, vis, txt, user_idx, item_idx, out);

    // Metapath: one block per batch row; output starts after the B scores.
    vtbpr_metapath_kernel<<<B, 128, 0, stream>>>(
        user_gama, item_gama, user_idx, item_idx, path_type, out + B);
}